// TorchWrithe_42614665511602
// MI455X (gfx1250) — compile-verified
//
#include <hip/hip_runtime.h>
#include <stdint.h>

#define N_ATOMS 128
#define N_SEG   7875      // segments: i in [0,124], j in [i+2,126]
#define PADLEN  8128      // padded triu length (pidx domain)
#define OUTW    16256     // 2 * PADLEN

struct V3 { float x, y, z; };

__device__ __forceinline__ V3 sub3(V3 a, V3 b) { return {a.x - b.x, a.y - b.y, a.z - b.z}; }
__device__ __forceinline__ float dot3(V3 a, V3 b) { return (a.x * b.x + a.y * b.y) + a.z * b.z; }
__device__ __forceinline__ V3 cross3(V3 a, V3 b) {
    return { a.y * b.z - a.z * b.y,
             a.z * b.x - a.x * b.z,
             a.x * b.y - a.y * b.x };
}
__device__ __forceinline__ V3 norm3(V3 a) {
    float n = sqrtf(dot3(a, a));
    return { a.x / n, a.y / n, a.z / n };
}
// g(a) = pidx(a, a+1) = first pidx of row a ; always an even product / 2
__device__ __forceinline__ int gfun(int a) { return (a * (255 - a)) >> 1; }
// segment linear index for segment (i, j)
__device__ __forceinline__ int segIdx(int i, int j) {
    return 125 * i - ((i * (i - 1)) >> 1) + j - i - 2;
}

__global__ void __launch_bounds__(256)
writhe_fused_kernel(const float* __restrict__ xyz,
                    const int*   __restrict__ segments,
                    const int*   __restrict__ sortIdx,
                    float*       __restrict__ out)
{
    __shared__ float s_xyz[N_ATOMS * 3];   // 1536 B
    __shared__ float s_wr[N_SEG];          // 31500 B

    const int b   = blockIdx.x;
    const int tid = threadIdx.x;
    const float* gx = xyz + (size_t)b * (N_ATOMS * 3);

    // ---- Stage coordinates to LDS with gfx1250 async global->LDS copy ----
    // 384 floats = 96 lanes x 16 bytes. Lanes >= 96 skip (EXEC-masked).
    if (tid < 96) {
        uint32_t lds_addr = (uint32_t)(uintptr_t)(&s_xyz[0]) + (uint32_t)tid * 16u;
        const float* gaddr = gx + tid * 4;
        asm volatile("global_load_async_to_lds_b128 %0, %1, off"
                     :: "v"(lds_addr), "v"(gaddr) : "memory");
    }
    asm volatile("s_wait_asynccnt 0" ::: "memory");
    __syncthreads();

    // ---- Phase A: writhe per segment ----
    for (int s = tid; s < N_SEG; s += 256) {
        int4 seg = ((const int4*)segments)[s];   // (i, i+1, j, j+1)
        int i = seg.x, j = seg.z;

        V3 p0 = { s_xyz[3*i],       s_xyz[3*i + 1],       s_xyz[3*i + 2]       };
        V3 p1 = { s_xyz[3*(i+1)],   s_xyz[3*(i+1) + 1],   s_xyz[3*(i+1) + 2]   };
        V3 p2 = { s_xyz[3*j],       s_xyz[3*j + 1],       s_xyz[3*j + 2]       };
        V3 p3 = { s_xyz[3*(j+1)],   s_xyz[3*(j+1) + 1],   s_xyz[3*(j+1) + 2]   };

        V3 d0 = norm3(sub3(p2, p0));
        V3 d1 = norm3(sub3(p3, p0));
        V3 d2 = norm3(sub3(p2, p1));
        V3 d3 = norm3(sub3(p3, p1));

        V3 axial = cross3(sub3(p3, p2), sub3(p1, p0));
        float sd = dot3(axial, d0);
        float sg = (sd > 0.0f) ? 1.0f : ((sd < 0.0f) ? -1.0f : 0.0f);

        V3 n0 = norm3(cross3(d0, d1));
        V3 n1 = norm3(cross3(d1, d3));
        V3 n2 = norm3(cross3(d3, d2));
        V3 n3 = norm3(cross3(d2, d0));

        float c0 = fminf(fmaxf(dot3(n0, n1), -1.0f), 1.0f);
        float c1 = fminf(fmaxf(dot3(n1, n2), -1.0f), 1.0f);
        float c2 = fminf(fmaxf(dot3(n2, n3), -1.0f), 1.0f);
        float c3 = fminf(fmaxf(dot3(n3, n0), -1.0f), 1.0f);

        float omega = ((asinf(c0) + asinf(c1)) + asinf(c2)) + asinf(c3);
        s_wr[s] = (omega * sg) / 6.28318530717958647692f;
    }
    __syncthreads();

    // ---- Phase B: inverse-scatter + pad + duplicate + permute, fused ----
    for (int k = tid; k < OUTW; k += 256) {
        int v = sortIdx[k];
        if (v >= PADLEN) v -= PADLEN;          // the two halves are identical
        float r = 0.0f;
        if (v != 0 && v != PADLEN - 1) {       // pad positions -> 0
            int P = v;                          // pidx(a,b) in [1, 8126]
            // decode row a: largest a with g(a) <= P ; float seed + fixup
            int a = (int)((255.0f - sqrtf((float)(65025 - 8 * P))) * 0.5f);
            if (a < 0)   a = 0;
            if (a > 126) a = 126;
            while (a < 126 && gfun(a + 1) <= P) ++a;
            while (a > 0  && gfun(a) > P)       --a;
            int bb = a + 1 + (P - gfun(a));     // column b

            // pair (a,bb) gathers wr from segments {a-1,a} x {bb-1,bb}
            #pragma unroll
            for (int t = 0; t < 4; ++t) {
                int i  = a  - (t >> 1);
                int jj = bb - (t & 1);
                if (i >= 0 && i <= 124 && jj >= i + 2 && jj <= 126)
                    r += s_wr[segIdx(i, jj)];
            }
        }
        out[(size_t)b * OUTW + k] = r;
    }
}

extern "C" void kernel_launch(void* const* d_in, const int* in_sizes, int n_in,
                              void* d_out, int out_size, void* d_ws, size_t ws_size,
                              hipStream_t stream) {
    const float* xyz      = (const float*)d_in[0];   // (B, 128, 3) f32
    const int*   segments = (const int*)d_in[1];     // (7875, 4) i32
    // d_in[2] = indices (unused: scatter inverted analytically)
    const int*   sortIdx  = (const int*)d_in[3];     // (16256,) i32
    float*       out      = (float*)d_out;           // (B, 16256) f32

    int B = in_sizes[0] / (N_ATOMS * 3);             // 512
    writhe_fused_kernel<<<B, 256, 0, stream>>>(xyz, segments, sortIdx, out);
}